// Attention_84516366450883
// MI455X (gfx1250) — compile-verified
//
#include <hip/hip_runtime.h>

typedef __attribute__((ext_vector_type(2))) float v2f;
typedef __attribute__((ext_vector_type(4))) float v4f;
typedef __attribute__((ext_vector_type(8))) float v8f;

#define EMB 128
#define NANC 8

__device__ __forceinline__ float fast_tanh(float x) {
#if __has_builtin(__builtin_amdgcn_tanhf)
    return __builtin_amdgcn_tanhf(x);
#else
    // tanh(x) = 1 - 2/(e^{2x}+1); saturates to +/-1 for large |x|
    float e = __expf(2.0f * x);
    return 1.0f - 2.0f * __builtin_amdgcn_rcpf(e + 1.0f);
#endif
}

__global__ __launch_bounds__(256)
void gram_attn_kernel(const float* __restrict__ W_emb,
                      const float* __restrict__ W_att,   // [256][128] row-major
                      const float* __restrict__ b_att,   // [128]
                      const float* __restrict__ v_att,   // [128]
                      const int*   __restrict__ leaves,  // [n_codes][8]
                      const int*   __restrict__ ancestors,
                      float* __restrict__ out,           // [n_codes][128]
                      int n_codes, int n_super, int total_waves)
{
    // B staged in WMMA-B-fragment pair layout: pair index k2*128+col holds
    // (W_att[2*k2][col], W_att[2*k2+1][col]) -> one ds_load_b64 per fragment.
    __shared__ v2f ldsB[128 * 128];   // 128 KB

    for (int i = threadIdx.x; i < 256 * 128; i += blockDim.x) {
        int k   = i >> 7;        // 0..255
        int col = i & 127;
        ((float*)ldsB)[(((k >> 1) * 128 + col) << 1) + (k & 1)] = W_att[i];
    }
    __syncthreads();

    const int lane = threadIdx.x & 31;
    const int c    = lane & 15;   // column-in-tile / A-row index
    const int h    = lane >> 4;   // half-of-wave
    const int wave = blockIdx.x * (blockDim.x >> 5) + (threadIdx.x >> 5);

    // bias and v columns for this lane's 8 column tiles (tile-invariant)
    float barr[8], varr[8];
#pragma unroll
    for (int ct = 0; ct < 8; ++ct) {
        barr[ct] = b_att[16 * ct + c];
        varr[ct] = v_att[16 * ct + c];
    }

    // Each wave handles a "supertile" of 4 codes = two 16-row WMMA A-tiles,
    // so every LDS B fragment feeds TWO wmma ops (halves LDS traffic/FLOP).
    for (int st = wave; st < n_super; st += total_waves) {
        int idxL[2], idxA[2];
#pragma unroll
        for (int t = 0; t < 2; ++t) {
            int rcode = st * 4 + 2 * t + (c >> 3);
            rcode = (rcode < n_codes) ? rcode : (n_codes - 1);
            const int ridx = rcode * NANC + (c & 7);
            idxL[t] = leaves[ridx];
            idxA[t] = ancestors[ridx];
        }

        v8f acc[2][8] = {};

        // K = 0..127 : leaf-embedding half of the concat
        {
            const float* a0 = W_emb + (size_t)idxL[0] * EMB;
            const float* a1 = W_emb + (size_t)idxL[1] * EMB;
#pragma unroll 4
            for (int ks = 0; ks < 32; ++ks) {
                v2f af0 = *(const v2f*)(a0 + 4 * ks + 2 * h);
                v2f af1 = *(const v2f*)(a1 + 4 * ks + 2 * h);
#pragma unroll
                for (int ct = 0; ct < 8; ++ct) {
                    v2f b = ldsB[(2 * ks + h) * 128 + 16 * ct + c];
                    acc[0][ct] = __builtin_amdgcn_wmma_f32_16x16x4_f32(
                        false, af0, false, b, (short)0, acc[0][ct], false, false);
                    acc[1][ct] = __builtin_amdgcn_wmma_f32_16x16x4_f32(
                        false, af1, false, b, (short)0, acc[1][ct], false, false);
                }
            }
        }
        // K = 128..255 : ancestor-embedding half
        {
            const float* a0 = W_emb + (size_t)idxA[0] * EMB;
            const float* a1 = W_emb + (size_t)idxA[1] * EMB;
#pragma unroll 4
            for (int ks = 32; ks < 64; ++ks) {
                v2f af0 = *(const v2f*)(a0 + 4 * (ks - 32) + 2 * h);
                v2f af1 = *(const v2f*)(a1 + 4 * (ks - 32) + 2 * h);
#pragma unroll
                for (int ct = 0; ct < 8; ++ct) {
                    v2f b = ldsB[(2 * ks + h) * 128 + 16 * ct + c];
                    acc[0][ct] = __builtin_amdgcn_wmma_f32_16x16x4_f32(
                        false, af0, false, b, (short)0, acc[0][ct], false, false);
                    acc[1][ct] = __builtin_amdgcn_wmma_f32_16x16x4_f32(
                        false, af1, false, b, (short)0, acc[1][ct], false, false);
                }
            }
        }

        // Epilogue per 16-row tile t: lane holds rows M = 8h+j, col 16ct+c
#pragma unroll
        for (int t = 0; t < 2; ++t) {
            float pre[8];
#pragma unroll
            for (int j = 0; j < 8; ++j) pre[j] = 0.0f;
#pragma unroll
            for (int ct = 0; ct < 8; ++ct) {
#pragma unroll
                for (int j = 0; j < 8; ++j) {
                    float m = fast_tanh(acc[t][ct][j] + barr[ct]);
                    pre[j] += m * varr[ct];
                }
            }
            // reduce over the 16 columns held by this half of the wave
#pragma unroll
            for (int m = 1; m <= 8; m <<= 1) {
#pragma unroll
                for (int j = 0; j < 8; ++j)
                    pre[j] += __shfl_xor(pre[j], m, 32);
            }

            // softmax over the 8 ancestors of code = st*4 + 2t + h (in-lane)
            float mx = pre[0];
#pragma unroll
            for (int j = 1; j < 8; ++j) mx = fmaxf(mx, pre[j]);
            float wgt[8], s = 0.0f;
#pragma unroll
            for (int j = 0; j < 8; ++j) { wgt[j] = __expf(pre[j] - mx); s += wgt[j]; }
            float inv = 1.0f / s;
#pragma unroll
            for (int j = 0; j < 8; ++j) wgt[j] *= inv;

            // attention-weighted sum of ancestor embeddings
            // lane covers cols [4c..4c+3] and [64+4c..64+4c+3]
            v4f o0 = {}, o1 = {};
#pragma unroll
            for (int a = 0; a < 8; ++a) {
                int ia = __shfl(idxA[t], 8 * h + a, 32); // anc idx for (code h, a)
                const float* r = W_emb + (size_t)ia * EMB;
                v4f r0 = *(const v4f*)(r + 4 * c);
                v4f r1 = *(const v4f*)(r + 64 + 4 * c);
                float wa = wgt[a];
#pragma unroll
                for (int j = 0; j < 4; ++j) {
                    o0[j] += wa * r0[j];
                    o1[j] += wa * r1[j];
                }
            }

            int code = st * 4 + 2 * t + h;
            if (code < n_codes) {
                float* op = out + (size_t)code * EMB;
                *(v4f*)(op + 4 * c)      = o0;
                *(v4f*)(op + 64 + 4 * c) = o1;
            }
        }
    }
}

extern "C" void kernel_launch(void* const* d_in, const int* in_sizes, int n_in,
                              void* d_out, int out_size, void* d_ws, size_t ws_size,
                              hipStream_t stream)
{
    const float* W_emb     = (const float*)d_in[0];
    const float* W_att     = (const float*)d_in[1];
    const float* b_att     = (const float*)d_in[2];
    const float* v_att     = (const float*)d_in[3];
    const int*   leaves    = (const int*)d_in[4];
    const int*   ancestors = (const int*)d_in[5];
    float*       out       = (float*)d_out;

    const int n_codes = in_sizes[4] / NANC;          // 100000
    const int n_super = (n_codes + 3) / 4;           // 4 codes per supertile

    int blocks = 512;                                // persistent-ish grid
    const int waves_per_blk = 256 / 32;
    int max_blocks = (n_super + waves_per_blk - 1) / waves_per_blk;
    if (blocks > max_blocks) blocks = max_blocks;
    const int total_waves = blocks * waves_per_blk;

    hipLaunchKernelGGL(gram_attn_kernel, dim3(blocks), dim3(256), 0, stream,
                       W_emb, W_att, b_att, v_att, leaves, ancestors, out,
                       n_codes, n_super, total_waves);
}